// PointGroup_25555055412090
// MI455X (gfx1250) — compile-verified
//
#include <hip/hip_runtime.h>
#include <cstdint>
#include <cstddef>

// ---------------------------------------------------------------------------
// Sparse 3D U-Net forward for MI455X (gfx1250, wave32).
// Core GEMMs use V_WMMA_F32_16X16X4_F32 (full fp32: the workload is
// L2-resident and gather-bound, nowhere near the matrix ceiling, so we keep
// reference precision). One wave per workgroup computes a 16x32 output tile.
// Conv kernel is templated so every hot loop is branch-free per variant.
// ---------------------------------------------------------------------------

typedef float v2f __attribute__((ext_vector_type(2)));
typedef float v8f __attribute__((ext_vector_type(8)));

#define BN_CHUNKS 32
#define LDS_ROW   392   // >= max (padded cin)+1 ; max cin = 2*192 = 384

// ---------------- fused gather(+BN+ReLU) sparse-conv WMMA kernel -----------
// y[r, dBase:dBase+32] = sum_tap sum_c act(x[nbr[r,tap], c]) * w[tap, c, d]
//   IDENT -> identity row mapping (dense GEMM, taps == 1)
//   ACT   -> pre-activation relu((x-mean)*istd) applied during the gather
//   ADD   -> y += addend (residual fuse)
//   PAD   -> cin not a multiple of 4 (input conv only, cin = 6)
template <bool ACT, bool IDENT, bool ADD, bool PAD>
__global__ __launch_bounds__(32) void conv_wmma_kernel(
    const float* __restrict__ xsrc,
    const int*   __restrict__ nbr,     // (n_out, taps) ; unused if IDENT
    const float* __restrict__ w,       // (taps, cin, cout)
    const float* __restrict__ meanv,   // (cin) ; unused if !ACT
    const float* __restrict__ istdv,   // (cin) ; unused if !ACT
    const float* __restrict__ addend,  // (n_out, cout) ; unused if !ADD
    float* __restrict__ y,             // (n_out, cout)
    int n_out, int cin, int cout, int taps)
{
    __shared__ float As[16 * LDS_ROW];
    __shared__ int   sIdx[16];

    const int lane    = threadIdx.x;     // 0..31 (wave32)
    const int rowBase = blockIdx.x << 4; // 16 voxel rows per wave
    const int dBase   = blockIdx.y << 5; // 32 output channels per wave
    const int cinp    = PAD ? ((cin + 3) & ~3) : cin;
    const int ldA     = cinp + 1;        // odd stride -> no LDS bank conflicts
    const int hi      = lane >> 4;       // 0 or 1 (half-wave)
    const int m       = lane & 15;

    v8f acc0 = {0.f, 0.f, 0.f, 0.f, 0.f, 0.f, 0.f, 0.f};
    v8f acc1 = acc0;

    for (int tap = 0; tap < taps; ++tap) {
        if (lane < 16) {
            int r = rowBase + lane;
            int idx = -1;
            if (r < n_out) idx = IDENT ? r : nbr[r * taps + tap];
            sIdx[lane] = idx;
        }
        __syncthreads();

        // Stage 16 gathered+activated rows into LDS (coalesced along channels,
        // no integer divides, branch-free per template variant)
        for (int r = 0; r < 16; ++r) {
            const int idx = sIdx[r];
            for (int c = lane; c < cinp; c += 32) {
                float v = 0.f;
                bool inb = idx >= 0;
                if (PAD) inb = inb && (c < cin);
                if (inb) {
                    v = xsrc[(size_t)idx * cin + c];
                    if (ACT) {
                        v = (v - meanv[c]) * istdv[c];
                        v = v > 0.f ? v : 0.f;
                    }
                }
                As[r * ldA + c] = v;
            }
        }
        __syncthreads();

        const float* wt = w + (size_t)tap * cin * cout;
        for (int c0 = 0; c0 < cinp; c0 += 4) {
            // A 16x4 f32 layout: lanes 0-15 hold K=0,1 ; lanes 16-31 hold K=2,3
            const int k0 = c0 + (hi << 1);
            v2f a;
            a.x = As[m * ldA + k0];
            a.y = As[m * ldA + k0 + 1];
            // B 4x16 f32: N = lane&15, K = 2*hi + e (mirrors C/D row split)
            const int col = dBase + m;
            v2f b0 = {0.f, 0.f};
            v2f b1 = {0.f, 0.f};
            if (!PAD || k0 < cin) {
                b0.x = wt[(size_t)k0 * cout + col];
                b1.x = wt[(size_t)k0 * cout + col + 16];
            }
            if (!PAD || k0 + 1 < cin) {
                b0.y = wt[(size_t)(k0 + 1) * cout + col];
                b1.y = wt[(size_t)(k0 + 1) * cout + col + 16];
            }
            acc0 = __builtin_amdgcn_wmma_f32_16x16x4_f32(
                false, a, false, b0, (short)0, acc0, false, false);
            acc1 = __builtin_amdgcn_wmma_f32_16x16x4_f32(
                false, a, false, b1, (short)0, acc1, false, false);
        }
        __syncthreads();
    }

    // D layout: element r -> M = r + 8*hi, N = lane&15
#pragma unroll
    for (int r = 0; r < 8; ++r) {
        int row = rowBase + (hi << 3) + r;
        if (row < n_out) {
            size_t o = (size_t)row * cout + dBase + m;
            float v0 = acc0[r], v1 = acc1[r];
            if (ADD) { v0 += addend[o]; v1 += addend[o + 16]; }
            y[o]      = v0;
            y[o + 16] = v1;
        }
    }
}

// ---------------- deterministic two-pass batch-norm statistics -------------
__global__ void bn_partial_kernel(const float* __restrict__ x, int n, int C,
                                  float* __restrict__ psum,
                                  float* __restrict__ psq)
{
    int c = blockIdx.y * blockDim.x + threadIdx.x;
    if (c >= C) return;
    int chunk = blockIdx.x;
    int per = (n + BN_CHUNKS - 1) / BN_CHUNKS;
    int r0 = chunk * per;
    int r1 = r0 + per; if (r1 > n) r1 = n;
    float s = 0.f, q = 0.f;
    for (int r = r0; r < r1; ++r) {
        float v = x[(size_t)r * C + c];
        s += v; q += v * v;
    }
    psum[chunk * C + c] = s;
    psq [chunk * C + c] = q;
}

__global__ void bn_finalize_kernel(const float* __restrict__ psum,
                                   const float* __restrict__ psq,
                                   int n, int C,
                                   float* __restrict__ meanv,
                                   float* __restrict__ istdv)
{
    int c = blockIdx.x * blockDim.x + threadIdx.x;
    if (c >= C) return;
    float s = 0.f, q = 0.f;
    for (int k = 0; k < BN_CHUNKS; ++k) { s += psum[k * C + c]; q += psq[k * C + c]; }
    float mu  = s / (float)n;
    float var = q / (float)n - mu * mu;
    meanv[c] = mu;
    istdv[c] = rsqrtf(var + 1e-4f);
}

// ---------------- upsample: u[r,d] = sum_c relu(bn(h[par[r],c])) * W[ko[r],c,d]
// writes into the second half of the concat buffer (row stride 2*Cd)
__global__ __launch_bounds__(64) void upsample_kernel(
    const float* __restrict__ h, const float* __restrict__ meanv,
    const float* __restrict__ istdv,
    const int* __restrict__ parents, const int* __restrict__ koffs,
    const float* __restrict__ upw,   // (8, Cs, Cd)
    float* __restrict__ cat,         // (n, 2*Cd)
    int n, int Cs, int Cd)
{
    int wave = (blockIdx.x * blockDim.x + threadIdx.x) >> 5; // one wave per row
    int lane = threadIdx.x & 31;
    if (wave >= n) return;
    int p  = parents[wave];
    int ko = koffs[wave];
    const float* wk = upw + (size_t)ko * Cs * Cd;
    for (int d = lane; d < Cd; d += 32) {
        float acc = 0.f;
        if (p >= 0) {
            for (int c = 0; c < Cs; ++c) {
                float v = (h[(size_t)p * Cs + c] - meanv[c]) * istdv[c];
                v = v > 0.f ? v : 0.f;
                acc += v * wk[(size_t)c * Cd + d];
            }
        }
        cat[(size_t)wave * (2 * Cd) + Cd + d] = acc;
    }
}

// ---------------- concat first half: cat[:, 0:C] = x -----------------------
__global__ void copy_cols_kernel(const float* __restrict__ x,
                                 float* __restrict__ cat, int n, int C)
{
    size_t i = (size_t)blockIdx.x * blockDim.x + threadIdx.x;
    size_t total = (size_t)n * C;
    if (i >= total) return;
    int r = (int)(i / C);
    int c = (int)(i - (size_t)r * C);
    cat[(size_t)r * (2 * C) + c] = x[i];
}

// ---------------- final: out[i,c] = bn(x)[input_map[i], c] -----------------
__global__ void final_gather_kernel(const float* __restrict__ x,
                                    const int* __restrict__ map,
                                    const float* __restrict__ meanv,
                                    const float* __restrict__ istdv,
                                    float* __restrict__ out, int npts, int C)
{
    size_t i = (size_t)blockIdx.x * blockDim.x + threadIdx.x;
    size_t total = (size_t)npts * C;
    if (i >= total) return;
    int r = (int)(i / C);
    int c = (int)(i - (size_t)r * C);
    out[i] = (x[(size_t)map[r] * C + c] - meanv[c]) * istdv[c];
}

// ---------------- host-side conv dispatch ----------------------------------
static inline void conv_launch(const float* x, const int* nb, const float* w,
                               const float* meanv, const float* istdv,
                               const float* add, float* y,
                               int nout, int cin, int cout, int taps,
                               hipStream_t stream)
{
    dim3 g((nout + 15) / 16, cout / 32);
    const bool act   = meanv != nullptr;
    const bool ident = nb == nullptr;
    const bool hadd  = add != nullptr;
    const bool pad   = (cin & 3) != 0;
    const int key = (act ? 8 : 0) | (ident ? 4 : 0) | (hadd ? 2 : 0) | (pad ? 1 : 0);
#define CW(A, I, D, P)                                                        \
    conv_wmma_kernel<A, I, D, P><<<g, 32, 0, stream>>>(                       \
        x, nb, w, meanv, istdv, add, y, nout, cin, cout, taps)
    switch (key) {
        case 0:  CW(false, false, false, false); break;
        case 1:  CW(false, false, false, true ); break;
        case 2:  CW(false, false, true,  false); break;
        case 3:  CW(false, false, true,  true ); break;
        case 4:  CW(false, true,  false, false); break;
        case 5:  CW(false, true,  false, true ); break;
        case 6:  CW(false, true,  true,  false); break;
        case 7:  CW(false, true,  true,  true ); break;
        case 8:  CW(true,  false, false, false); break;
        case 9:  CW(true,  false, false, true ); break;
        case 10: CW(true,  false, true,  false); break;
        case 11: CW(true,  false, true,  true ); break;
        case 12: CW(true,  true,  false, false); break;
        case 13: CW(true,  true,  false, true ); break;
        case 14: CW(true,  true,  true,  false); break;
        default: CW(true,  true,  true,  true ); break;
    }
#undef CW
}

// ===========================================================================
extern "C" void kernel_launch(void* const* d_in, const int* in_sizes, int n_in,
                              void* d_out, int out_size, void* d_ws, size_t ws_size,
                              hipStream_t stream)
{
    const int C[7] = {32, 64, 96, 128, 160, 192, 224};
    int n[7];
    for (int l = 0; l < 7; ++l) n[l] = in_sizes[2 + l] / 27;

    const float* feats     = (const float*)d_in[0];
    const int*   input_map = (const int*)  d_in[1];
    const int* nbr[7];   for (int l = 0; l < 7; ++l) nbr[l]   = (const int*)d_in[2 + l];
    const int* down[6];  for (int l = 0; l < 6; ++l) down[l]  = (const int*)d_in[9 + l];
    const int* upar[6];  for (int l = 0; l < 6; ++l) upar[l]  = (const int*)d_in[15 + l];
    const int* ukoff[6]; for (int l = 0; l < 6; ++l) ukoff[l] = (const int*)d_in[21 + l];

    // ---- parse params; auto-detect insertion-order vs jax-sorted flattening
    int p = 27;
    const float* input_w = (const float*)d_in[p++];
    struct Lv {
        const float *b0w1, *b0w2, *b1w1, *b1w2;
        const float *down_w, *up_w, *t0w1, *t0w2, *t0iw, *t1w1, *t1w2;
    } L[7];
    for (int l = 0; l < 7; ++l) {
        L[l].b0w1 = (const float*)d_in[p++];
        L[l].b0w2 = (const float*)d_in[p++];
        L[l].b1w1 = (const float*)d_in[p++];
        L[l].b1w2 = (const float*)d_in[p++];
        L[l].down_w = L[l].up_w = L[l].t0w1 = L[l].t0w2 = L[l].t0iw =
            L[l].t1w1 = L[l].t1w2 = nullptr;
        if (l < 6) {
            L[l].down_w = (const float*)d_in[p];
            // next size: 8*C*C' => insertion order (up_w next)
            //            2*C*C  => sorted keys (tail[0].iw next)
            bool sorted = (in_sizes[p + 1] == 2 * C[l] * C[l]);
            ++p;
            if (!sorted) {
                L[l].up_w = (const float*)d_in[p++];
                L[l].t0w1 = (const float*)d_in[p++];
                L[l].t0w2 = (const float*)d_in[p++];
                L[l].t0iw = (const float*)d_in[p++];
                L[l].t1w1 = (const float*)d_in[p++];
                L[l].t1w2 = (const float*)d_in[p++];
            } else {
                L[l].t0iw = (const float*)d_in[p++];
                L[l].t0w1 = (const float*)d_in[p++];
                L[l].t0w2 = (const float*)d_in[p++];
                L[l].t1w1 = (const float*)d_in[p++];
                L[l].t1w2 = (const float*)d_in[p++];
                L[l].up_w = (const float*)d_in[p++];
            }
        }
    }

    // ---- workspace bump allocator (floats) --------------------------------
    float* wsf = (float*)d_ws;
    size_t off = 0;
    auto alloc = [&](size_t e) { size_t o = off; off += (e + 63) & ~(size_t)63; return o; };

    float* X[7];
    for (int l = 0; l < 7; ++l) X[l] = wsf + alloc((size_t)n[l] * C[l]);
    size_t maxNC = 0, maxN2C = 0;
    for (int l = 0; l < 7; ++l) { size_t a = (size_t)n[l] * C[l]; if (a > maxNC) maxNC = a; }
    for (int l = 0; l < 6; ++l) { size_t a = (size_t)n[l] * 2 * C[l]; if (a > maxN2C) maxN2C = a; }
    float* H     = wsf + alloc(maxNC);
    float* G     = wsf + alloc(maxNC);
    float* I     = wsf + alloc(maxNC);
    float* CAT   = wsf + alloc(maxN2C);
    float* psum  = wsf + alloc((size_t)BN_CHUNKS * 512);
    float* psq   = wsf + alloc((size_t)BN_CHUNKS * 512);
    float* meanB = wsf + alloc(512);
    float* istdB = wsf + alloc(512);
    (void)ws_size; (void)n_in;

    // ---- launch helpers ----------------------------------------------------
    auto bn = [&](const float* x, int nn, int cc) {
        dim3 g(BN_CHUNKS, (cc + 127) / 128);
        bn_partial_kernel<<<g, 128, 0, stream>>>(x, nn, cc, psum, psq);
        bn_finalize_kernel<<<(cc + 127) / 128, 128, 0, stream>>>(psum, psq, nn, cc, meanB, istdB);
    };
    auto conv = [&](const float* x, const int* nb, const float* w, bool act,
                    const float* add, float* yy, int nout, int cinv, int coutv, int taps) {
        conv_launch(x, nb, w,
                    act ? (const float*)meanB : (const float*)nullptr,
                    act ? (const float*)istdB : (const float*)nullptr,
                    add, yy, nout, cinv, coutv, taps, stream);
    };
    auto resblock = [&](const float* xin, float* outb, int nn, int cinv, int cv,
                        const int* nb, const float* w1, const float* w2, const float* iw) {
        const float* idt = xin;
        if (iw) { conv(xin, nullptr, iw, false, nullptr, I, nn, cinv, cv, 1); idt = I; }
        bn(xin, nn, cinv);
        conv(xin, nb, w1, true, nullptr, H, nn, cinv, cv, 27);
        bn(H, nn, cv);
        conv(H, nb, w2, true, idt, outb, nn, cv, cv, 27);
    };

    // ---- input conv (cin=6 padded to 8 inside the kernel) ------------------
    conv(feats, nbr[0], input_w, false, nullptr, X[0], n[0], 6, 32, 27);

    // ---- encoder sweep -----------------------------------------------------
    for (int l = 0; l < 7; ++l) {
        resblock(X[l], G, n[l], C[l], C[l], nbr[l], L[l].b0w1, L[l].b0w2, nullptr);
        resblock(G, X[l], n[l], C[l], C[l], nbr[l], L[l].b1w1, L[l].b1w2, nullptr);
        if (l < 6) {
            bn(X[l], n[l], C[l]);
            conv(X[l], down[l], L[l].down_w, true, nullptr,
                 X[l + 1], n[l + 1], C[l], C[l + 1], 8);
        }
    }

    // ---- decoder sweep -----------------------------------------------------
    for (int l = 5; l >= 0; --l) {
        bn(X[l + 1], n[l + 1], C[l + 1]);
        int nblk = (n[l] + 1) / 2;  // 2 waves (rows) per 64-thread block
        upsample_kernel<<<nblk, 64, 0, stream>>>(
            X[l + 1], meanB, istdB, upar[l], ukoff[l], L[l].up_w,
            CAT, n[l], C[l + 1], C[l]);
        size_t tot = (size_t)n[l] * C[l];
        copy_cols_kernel<<<(unsigned)((tot + 255) / 256), 256, 0, stream>>>(
            X[l], CAT, n[l], C[l]);
        resblock(CAT, G, n[l], 2 * C[l], C[l], nbr[l],
                 L[l].t0w1, L[l].t0w2, L[l].t0iw);
        resblock(G, X[l], n[l], C[l], C[l], nbr[l],
                 L[l].t1w1, L[l].t1w2, nullptr);
    }

    // ---- final BN + gather -------------------------------------------------
    bn(X[0], n[0], 32);
    int npts = in_sizes[1];
    size_t tot = (size_t)npts * 32;
    final_gather_kernel<<<(unsigned)((tot + 255) / 256), 256, 0, stream>>>(
        X[0], input_map, meanB, istdB, (float*)d_out, npts, 32);
    (void)out_size;
}